// FuzzyPooling_1769526526236
// MI455X (gfx1250) — compile-verified
//
#include <hip/hip_runtime.h>
#include <cstdint>

#define AS1 __attribute__((address_space(1)))
#define AS3 __attribute__((address_space(3)))

typedef __attribute__((ext_vector_type(4))) int v4i;

#if __has_builtin(__builtin_amdgcn_global_load_async_to_lds_b128)
#define FUZZY_ASYNC_LDS 1
#else
#define FUZZY_ASYNC_LDS 0
#endif

namespace {
constexpr int kB = 4, kH = 384, kW = 384, kC = 48;
constexpr int kR = kH / 3;    // 128 output rows
constexpr int kCc = kW / 3;   // 128 output cols
constexpr int CCT = 32;       // output cols per workgroup tile
constexpr int TPB = 256;
constexpr int SEG_FLOATS = CCT * 3 * kC;  // 4608 floats per staged input-row segment
constexpr int SEG_VEC4 = SEG_FLOATS / 4;  // 1152
constexpr int TOTAL_VEC4 = 3 * SEG_VEC4;  // 3456
constexpr int OUT_PER_WG = CCT * kC;      // 1536 outputs per workgroup
constexpr float kEps = 1e-4f;
constexpr float kNegHalfLog2e = -0.72134752044448169036f;  // -0.5 * log2(e)
}  // namespace

__device__ __forceinline__ float fast_exp2(float v) {
#if __has_builtin(__builtin_amdgcn_exp2f)
  return __builtin_amdgcn_exp2f(v);  // raw v_exp_f32
#else
  return exp2f(v);
#endif
}

__device__ __forceinline__ float fast_rcp(float v) {
#if __has_builtin(__builtin_amdgcn_rcpf)
  return __builtin_amdgcn_rcpf(v);  // raw v_rcp_f32
#else
  return 1.0f / v;
#endif
}

// One 9-tuple -> pooled scalar (faithful to reference fuzzy_pool math).
__device__ __forceinline__ float fuzzy_pool9(const float x9[9]) {
  // Nested slice sums: [3:6] ⊂ [2:7] ⊂ [1:8] ⊂ [0:9]
  const float s3 = x9[3] + x9[4] + x9[5];
  const float s2 = s3 + x9[2] + x9[6];
  const float s1 = s2 + x9[1] + x9[7];
  const float s0 = s1 + x9[0] + x9[8];

  float kmm[5];
  kmm[0] = s1 * (1.0f / 7.0f);
  kmm[1] = s2 * (1.0f / 5.0f);
  kmm[2] = s3 * (1.0f / 3.0f);
  kmm[3] = x9[4];
  kmm[4] = s0 * (1.0f / 9.0f);

  const float v_avg = (kmm[0] + kmm[1] + kmm[2] + kmm[3] + kmm[4]) * 0.2f;

  float o[9];
#pragma unroll
  for (int t = 0; t < 9; ++t) o[t] = fabsf(x9[t] - v_avg);

  const float u3 = o[3] + o[4] + o[5];
  const float u2 = u3 + o[2] + o[6];
  const float u1 = u2 + o[1] + o[7];
  const float u0 = u1 + o[0] + o[8];

  float var[5];
  var[0] = u1 * (1.0f / 7.0f) + kEps;
  var[1] = u2 * (1.0f / 5.0f) + kEps;
  var[2] = u3 * (1.0f / 3.0f) + kEps;
  var[3] = o[4] + kEps;
  var[4] = u0 * (1.0f / 9.0f) + kEps;

  float acc[9], mx[9];
#pragma unroll
  for (int t = 0; t < 9; ++t) {
    acc[t] = 0.0f;
    mx[t] = 0.0f;  // pi >= 0 always
  }

  // pi[j][t] = exp(-0.5*(x_t - kmm_j)^2 / var_j) = exp2(sj * d^2)
#pragma unroll
  for (int j = 0; j < 5; ++j) {
    const float sj = kNegHalfLog2e * fast_rcp(var[j]);
#pragma unroll
    for (int t = 0; t < 9; ++t) {
      const float d = x9[t] - kmm[j];
      const float p = fast_exp2(sj * d * d);
      acc[t] += p;
      mx[t] = fmaxf(mx[t], p);
    }
  }

  float thresh = mx[0];
#pragma unroll
  for (int t = 1; t < 9; ++t) thresh = fminf(thresh, mx[t]);

  bool m_mem = false;
  float num = 0.0f, den = 0.0f;
#pragma unroll
  for (int t = 0; t < 9; ++t) {
    const float a = acc[t] * 0.2f;  // avg_pi[t]
    m_mem = m_mem || (a > thresh);
    num += a * x9[t];
    den += a;
  }

  float pooled;
  if (m_mem) {
    pooled = kmm[4];  // mean_x
  } else if (var[4] < kEps) {
    pooled = v_avg;
  } else {
    pooled = num / den;  // denoised
  }
  return pooled;
}

__global__ __launch_bounds__(TPB) void fuzzy_pool_kernel(const float* __restrict__ x,
                                                         float* __restrict__ out) {
  __shared__ __align__(16) float lds[3 * SEG_FLOATS];  // 54 KB

  const int wg = blockIdx.x;          // [0, 4*128*4)
  const int cc0 = (wg & 3) * CCT;     // output-col tile start
  const int r = (wg >> 2) & (kR - 1); // output row
  const int b = wg >> 9;              // batch
  const int tid = threadIdx.x;

  // Three contiguous global row segments: row = 3r+kr, float offset 144*cc0, len 4608.
  const size_t gbase = ((size_t)(b * kH + 3 * r) * kW + 3 * cc0) * kC;
  const size_t growstride = (size_t)kW * kC;

  for (int i = tid; i < TOTAL_VEC4; i += TPB) {
    const int seg = i / SEG_VEC4;
    const int off4 = i - seg * SEG_VEC4;
    const float* gp = x + gbase + (size_t)seg * growstride + (size_t)off4 * 4;
    float* lp = &lds[seg * SEG_FLOATS + off4 * 4];
#if FUZZY_ASYNC_LDS
    __builtin_amdgcn_global_load_async_to_lds_b128(
        (AS1 v4i*)(uintptr_t)gp,
        (AS3 v4i*)(uint32_t)(uintptr_t)lp,
        0, 0);
#else
    *reinterpret_cast<float4*>(lp) = *reinterpret_cast<const float4*>(gp);
#endif
  }

#if FUZZY_ASYNC_LDS
#if __has_builtin(__builtin_amdgcn_s_wait_asynccnt)
  __builtin_amdgcn_s_wait_asynccnt(0);
#else
  asm volatile("s_wait_asynccnt 0" ::: "memory");
#endif
#endif
  __syncthreads();

  // Each output (cc_local, q) consumes 9 consecutive LDS floats at
  // 144*cc_local + 4464*kr + 9*q, with kr = q/16 (constant per tuple).
  const size_t outbase = ((size_t)(b * kR + r) * kCc + cc0) * kC;
#pragma unroll
  for (int k = 0; k < OUT_PER_WG / TPB; ++k) {
    const int idx = tid + k * TPB;
    const int cc_local = idx / kC;
    const int q = idx - cc_local * kC;
    const int kr = q >> 4;
    const int lbase = cc_local * 144 + kr * 4464 + q * 9;

    float x9[9];
#pragma unroll
    for (int t = 0; t < 9; ++t) x9[t] = lds[lbase + t];

    out[outbase + idx] = fuzzy_pool9(x9);
  }
}

extern "C" void kernel_launch(void* const* d_in, const int* in_sizes, int n_in,
                              void* d_out, int out_size, void* d_ws, size_t ws_size,
                              hipStream_t stream) {
  (void)in_sizes; (void)n_in; (void)out_size; (void)d_ws; (void)ws_size;
  const float* x = (const float*)d_in[0];
  float* out = (float*)d_out;
  const dim3 grid(kB * kR * (kCc / CCT));  // 2048 workgroups
  const dim3 block(TPB);
  hipLaunchKernelGGL(fuzzy_pool_kernel, grid, block, 0, stream, x, out);
}